// Tree_SSM_49847390437471
// MI455X (gfx1250) — compile-verified
//
#include <hip/hip_runtime.h>
#include <hip/hip_bf16.h>
#include <math.h>

// Problem dims (fixed by reference): L=4096, D=256, N=16, batch=1.
#define LSEQ 4096
#define DDIM 256
#define NDIM 16
#define NCHUNK 64           // chunks along L
#define CHUNK  64           // L per chunk (NCHUNK*CHUNK == LSEQ)

typedef __attribute__((ext_vector_type(2))) float v2f;
typedef __attribute__((ext_vector_type(8))) float v8f;

// ---------------------------------------------------------------------------
// Kernel 1: fp32 WMMA projections.
//   job  0..15 : delta tile  -> softplus(X@W1 + b1)   out [L, D]
//   job   16   : Bm          -> X@W2 + b2             out [L, N]
//   job   17   : Cm          -> X@W3 + b3             out [L, N]
// One wave (32 lanes) computes one 16x16 output tile, K=256 via 64 x
// V_WMMA_F32_16X16X4_F32 (two interleaved accumulators for ILP).
// ---------------------------------------------------------------------------
__global__ __launch_bounds__(32) void proj_wmma_kernel(
    const float* __restrict__ X,
    const float* __restrict__ W1, const float* __restrict__ bias1,
    const float* __restrict__ W2, const float* __restrict__ bias2,
    const float* __restrict__ W3, const float* __restrict__ bias3,
    float* __restrict__ deltaO, float* __restrict__ BmO, float* __restrict__ CmO)
{
    const int t    = threadIdx.x;     // 0..31
    const int m16  = t & 15;
    const int kh   = t >> 4;          // 0 or 1 (which K-half this lane holds)
    const int rt   = blockIdx.x;      // row tile 0..255
    const int job  = blockIdx.y;      // 0..17

    const float* W; const float* bias; float* out; int ncols; int ctile; bool do_sp;
    if (job < 16)       { W = W1; bias = bias1; out = deltaO; ncols = DDIM; ctile = job; do_sp = true;  }
    else if (job == 16) { W = W2; bias = bias2; out = BmO;    ncols = NDIM; ctile = 0;   do_sp = false; }
    else                { W = W3; bias = bias3; out = CmO;    ncols = NDIM; ctile = 0;   do_sp = false; }

    const int row = rt * 16 + m16;          // A-matrix row owned by this lane
    const int col = ctile * 16 + m16;       // B/D column owned by this lane
    const float* xrow = X + (size_t)row * DDIM;

    v8f acc0 = {};
    v8f acc1 = {};
    #pragma unroll 4
    for (int k = 0; k < DDIM; k += 8) {
        // K-slab [k, k+4): lane holds A[row][k+2*kh .. +1] (aligned b64 load)
        v2f a0 = *(const v2f*)(xrow + k + 2 * kh);
        v2f b0;
        b0.x = W[(size_t)(k + 2 * kh)     * ncols + col];
        b0.y = W[(size_t)(k + 2 * kh + 1) * ncols + col];
        acc0 = __builtin_amdgcn_wmma_f32_16x16x4_f32(
            false, a0, false, b0, (short)0, acc0, false, false);
        // K-slab [k+4, k+8): independent accumulator -> no WMMA RAW stall
        v2f a1 = *(const v2f*)(xrow + k + 4 + 2 * kh);
        v2f b1;
        b1.x = W[(size_t)(k + 4 + 2 * kh)     * ncols + col];
        b1.y = W[(size_t)(k + 4 + 2 * kh + 1) * ncols + col];
        acc1 = __builtin_amdgcn_wmma_f32_16x16x4_f32(
            false, a1, false, b1, (short)0, acc1, false, false);
    }
    v8f acc = acc0 + acc1;

    const float bv = bias[col];
    #pragma unroll
    for (int r = 0; r < 8; ++r) {
        const int m = rt * 16 + r + 8 * kh;   // D-matrix row for VGPR r
        float v = acc[r] + bv;
        if (do_sp) v = (v > 20.0f) ? v : log1pf(__expf(v));  // softplus
        out[(size_t)m * ncols + col] = v;
    }
}

// ---------------------------------------------------------------------------
// Kernel 2: phase A of chunked scan. Block g handles chunk g; thread d owns
// channel d with all 16 state dims in registers. Produces, per (g,d,n):
//   P = prod_{l in chunk} exp(delta*A)          (chunk transition)
//   S = local scan final state with h=0 entry   (chunk contribution)
// ---------------------------------------------------------------------------
__global__ __launch_bounds__(256) void scan_phaseA_kernel(
    const float* __restrict__ delta, const float* __restrict__ Bm,
    const float* __restrict__ X,     const float* __restrict__ Amat,
    float* __restrict__ P, float* __restrict__ S)
{
    __shared__ float sB[CHUNK * NDIM];
    const int g = blockIdx.x;
    const int d = threadIdx.x;

    #pragma unroll
    for (int i = d; i < CHUNK * NDIM; i += 256) sB[i] = Bm[(size_t)g * CHUNK * NDIM + i];
    __syncthreads();

    float Ar[NDIM], h[NDIM], p[NDIM];
    #pragma unroll
    for (int n = 0; n < NDIM; ++n) {
        Ar[n] = Amat[d * NDIM + n];
        h[n] = 0.0f;
        p[n] = 1.0f;
    }

    for (int j = 0; j < CHUNK; ++j) {
        const int l = g * CHUNK + j;
        const float dl = delta[(size_t)l * DDIM + d];
        const float xv = X[(size_t)l * DDIM + d];
        const float dx = dl * xv;
        #pragma unroll
        for (int n = 0; n < NDIM; ++n) {
            const float da = __expf(dl * Ar[n]);
            h[n] = da * h[n] + dx * sB[j * NDIM + n];
            p[n] *= da;
        }
    }

    const size_t base = ((size_t)g * DDIM + d) * NDIM;
    #pragma unroll
    for (int n = 0; n < NDIM; ++n) { P[base + n] = p[n]; S[base + n] = h[n]; }
}

// ---------------------------------------------------------------------------
// Kernel 3: phase B — sequential combine across the 64 chunks, one thread per
// (d,n) state element (4096 threads). H0[g] = state entering chunk g.
// ---------------------------------------------------------------------------
__global__ __launch_bounds__(256) void scan_phaseB_kernel(
    const float* __restrict__ P, const float* __restrict__ S, float* __restrict__ H0)
{
    const int idx = blockIdx.x * 256 + threadIdx.x;   // idx = d*16 + n
    float carry = 0.0f;
    for (int g = 0; g < NCHUNK; ++g) {
        const size_t o = (size_t)g * (DDIM * NDIM) + idx;
        H0[o] = carry;
        carry = P[o] * carry + S[o];
    }
}

// ---------------------------------------------------------------------------
// Kernel 4: phase C — re-run each chunk from its true entry state and emit
// y[l,d] = sum_n C[l,n] * h[d,n] (reduction stays inside the thread).
// ---------------------------------------------------------------------------
__global__ __launch_bounds__(256) void scan_phaseC_kernel(
    const float* __restrict__ delta, const float* __restrict__ Bm,
    const float* __restrict__ Cm,    const float* __restrict__ X,
    const float* __restrict__ Amat,  const float* __restrict__ H0,
    float* __restrict__ Y)
{
    __shared__ float sB[CHUNK * NDIM];
    __shared__ float sC[CHUNK * NDIM];
    const int g = blockIdx.x;
    const int d = threadIdx.x;

    #pragma unroll
    for (int i = d; i < CHUNK * NDIM; i += 256) {
        sB[i] = Bm[(size_t)g * CHUNK * NDIM + i];
        sC[i] = Cm[(size_t)g * CHUNK * NDIM + i];
    }
    __syncthreads();

    float Ar[NDIM], h[NDIM];
    const size_t base = ((size_t)g * DDIM + d) * NDIM;
    #pragma unroll
    for (int n = 0; n < NDIM; ++n) {
        Ar[n] = Amat[d * NDIM + n];
        h[n]  = H0[base + n];
    }

    for (int j = 0; j < CHUNK; ++j) {
        const int l = g * CHUNK + j;
        const float dl = delta[(size_t)l * DDIM + d];
        const float xv = X[(size_t)l * DDIM + d];
        const float dx = dl * xv;
        float y = 0.0f;
        #pragma unroll
        for (int n = 0; n < NDIM; ++n) {
            const float da = __expf(dl * Ar[n]);
            h[n] = da * h[n] + dx * sB[j * NDIM + n];
            y += sC[j * NDIM + n] * h[n];
        }
        Y[(size_t)l * DDIM + d] = y;
    }
}

// ---------------------------------------------------------------------------
extern "C" void kernel_launch(void* const* d_in, const int* in_sizes, int n_in,
                              void* d_out, int out_size, void* d_ws, size_t ws_size,
                              hipStream_t stream) {
    const float* X   = (const float*)d_in[0];   // [1,1,L,D] -> [L,D]
    // d_in[1] = adj (unused by reference)
    const float* W1  = (const float*)d_in[2];   // [D,D]
    const float* bb1 = (const float*)d_in[3];   // [D]
    const float* W2  = (const float*)d_in[4];   // [D,N]
    const float* bb2 = (const float*)d_in[5];   // [N]
    const float* W3  = (const float*)d_in[6];   // [D,N]
    const float* bb3 = (const float*)d_in[7];   // [N]
    const float* A   = (const float*)d_in[8];   // [D,N]
    float* Y = (float*)d_out;                   // [1,L,D]

    float* ws    = (float*)d_ws;
    float* delta = ws;                                   // L*D       = 1048576
    float* Bm    = delta + (size_t)LSEQ * DDIM;          // L*N       =   65536
    float* Cm    = Bm    + (size_t)LSEQ * NDIM;          // L*N       =   65536
    float* P     = Cm    + (size_t)LSEQ * NDIM;          // G*D*N     =  262144
    float* S     = P     + (size_t)NCHUNK * DDIM * NDIM; // G*D*N
    float* H0    = S     + (size_t)NCHUNK * DDIM * NDIM; // G*D*N

    dim3 gproj(LSEQ / 16, 18);   // 16 delta col-tiles + Bm + Cm
    proj_wmma_kernel<<<gproj, 32, 0, stream>>>(X, W1, bb1, W2, bb2, W3, bb3,
                                               delta, Bm, Cm);
    scan_phaseA_kernel<<<NCHUNK, DDIM, 0, stream>>>(delta, Bm, X, A, P, S);
    scan_phaseB_kernel<<<(DDIM * NDIM) / 256, 256, 0, stream>>>(P, S, H0);
    scan_phaseC_kernel<<<NCHUNK, DDIM, 0, stream>>>(delta, Bm, Cm, X, A, H0, Y);
}